// SVS_21122649161804
// MI455X (gfx1250) — compile-verified
//
#include <hip/hip_runtime.h>

// MI455X / gfx1250, wave32. fp32 GAT forward using V_WMMA_F32_16X16X4_F32 for
// every matmul, with async global->LDS staging (ASYNCcnt path) when available.

typedef __attribute__((ext_vector_type(2))) float v2f;
typedef __attribute__((ext_vector_type(4))) float v4f;
typedef __attribute__((ext_vector_type(8))) float v8f;
typedef int v4i_vs __attribute__((vector_size(16)));   // matches builtin proto

#define B_   128
#define N_   256
#define F_   36
#define D_   128
#define H_   8
#define DK_  16
#define L_   4
#define FC_  256
#define NEG_ (-9.0e15f)

#if defined(__has_builtin)
# if __has_builtin(__builtin_amdgcn_global_load_async_to_lds_b128) && \
     __has_builtin(__builtin_amdgcn_s_wait_asynccnt)
#  define USE_ASYNC_LDS 1
# endif
#endif

// 16-byte async copy global -> LDS (falls back to a plain vec4 staging copy)
__device__ __forceinline__ void copy16_g2lds(const float* gsrc, float* ldst)
{
#ifdef USE_ASYNC_LDS
    __builtin_amdgcn_global_load_async_to_lds_b128(
        (__attribute__((address_space(1))) v4i_vs*)const_cast<float*>(gsrc),
        (__attribute__((address_space(3))) v4i_vs*)ldst,
        0, 0);
#else
    *(v4f*)ldst = *(const v4f*)gsrc;
#endif
}
__device__ __forceinline__ void async_wait_all()
{
#ifdef USE_ASYNC_LDS
    __builtin_amdgcn_s_wait_asynccnt(0);
#endif
}

// ---------------------------------------------------------------------------
// Generic fp32 WMMA GEMM: C[M,Nc] = act(A[M,K] @ W[K,Nc] + bias)
// One 16x16 output tile per wave, 8 waves per block, tiles ordered tn-fastest.
// Requires M%16==0, Nc%16==0, K%4==0, K<=256, (Nc/16)%8==0 (so the 8 waves of
// a block share one 16xK A slab, staged once in LDS). The slab is a fully
// contiguous 16*K-float region of A, DMA'd via async-to-LDS.
// ---------------------------------------------------------------------------
__global__ __launch_bounds__(256)
void gemm_f32_wmma(const float* __restrict__ A, const float* __restrict__ W,
                   const float* __restrict__ bias, float* __restrict__ C,
                   int M, int K, int Nc, int relu)
{
    __shared__ __align__(16) float As[16 * 256];
    const int lane = threadIdx.x & 31;
    const int wave = threadIdx.x >> 5;
    const int tilesPerRow = Nc >> 4;
    const int tile0 = blockIdx.x * 8;
    const int tm   = tile0 / tilesPerRow;     // uniform across the block
    const int row0 = tm * 16;

    // stage the contiguous 16 x K A-slab into LDS (async DMA, 16B granules)
    const float* slab = A + (size_t)row0 * K;
    for (int idx = threadIdx.x; idx < 4 * K; idx += 256)
        copy16_g2lds(slab + idx * 4, As + idx * 4);
    async_wait_all();
    __syncthreads();

    const int tile = tile0 + wave;
    const int tn   = tile % tilesPerRow;
    const int col0 = tn * 16;
    const int half = lane >> 4;               // lanes 0-15 vs 16-31
    const int l16  = lane & 15;

    v8f acc = {};
    for (int k0 = 0; k0 < K; k0 += 4) {
        v2f a, b;
        // A frag (16x4 f32): lane row = l16; VGPR0/1 hold K = k0+2*half (+1)
        a[0] = As[l16 * K + k0 + 2 * half];
        a[1] = As[l16 * K + k0 + 2 * half + 1];
        // B frag (4x16 f32): row K = k0+2*half (+1), col = l16
        b[0] = W[(size_t)(k0 + 2 * half)     * Nc + col0 + l16];
        b[1] = W[(size_t)(k0 + 2 * half + 1) * Nc + col0 + l16];
        // prefetch next step's W rows into cache
        __builtin_prefetch(&W[(size_t)(k0 + 4 + 2 * half) * Nc + col0 + l16], 0, 1);
        acc = __builtin_amdgcn_wmma_f32_16x16x4_f32(false, a, false, b,
                                                    (short)0, acc, false, false);
    }

    const int   col = col0 + l16;
    const float bv  = bias ? bias[col] : 0.0f;
    #pragma unroll
    for (int r = 0; r < 8; ++r) {
        int   row = row0 + r + 8 * half;      // C/D: VGPR r = rows r / r+8
        float v   = acc[r] + bv;
        if (relu) v = fmaxf(v, 0.0f);
        C[(size_t)row * Nc + col] = v;
    }
}

// ---------------------------------------------------------------------------
// Per-head attention logits inputs: src/dst[b,h,n] = <hw[b,n,h,:], a_{src/dst}[h,:]>
// ---------------------------------------------------------------------------
__global__ __launch_bounds__(256)
void gat_srcdst(const float* __restrict__ hw, const float* __restrict__ asrc,
                const float* __restrict__ adst,
                float* __restrict__ src, float* __restrict__ dst)
{
    int tid  = blockIdx.x * blockDim.x + threadIdx.x;   // B*H*N
    int n    = tid % N_;
    int bh   = tid / N_;
    int head = bh % H_;
    int b    = bh / H_;
    const float* hp = hw + (size_t)(b * N_ + n) * D_ + head * DK_;
    const float* s  = asrc + head * DK_;
    const float* t  = adst + head * DK_;
    float s1 = 0.f, s2 = 0.f;
    #pragma unroll
    for (int d = 0; d < DK_; ++d) { float v = hp[d]; s1 += v * s[d]; s2 += v * t[d]; }
    src[tid] = s1;                                      // [B,H,N]
    dst[tid] = s2;
}

// ---------------------------------------------------------------------------
// Fused masked softmax + WMMA aggregation + residual + ReLU (in-place on h).
// One block per (b, head, 16-row i-tile).
//   async: DMA hw[b,:,head,:] (256x16, 16KB) into LDS, overlapping phase 1
//   phase 1: P[i][j] = exp(leakyrelu(src_i+dst_j) masked - rowmax) into LDS
//   phase 2: out = P[16,256] @ hw_s[256,16]  (K split across 8 waves, WMMA)
//   then: h = relu(out/rowsum + h)
// ---------------------------------------------------------------------------
__global__ __launch_bounds__(256)
void gat_attn(const float* __restrict__ Aadj, const float* __restrict__ hw,
              const float* __restrict__ src, const float* __restrict__ dst,
              float* __restrict__ h)
{
    __shared__ __align__(16) float p_s[16 * N_];   // unnormalized probs, 16 KB
    __shared__ __align__(16) float hw_s[N_ * 16];  // B-operand slab, 16 KB
    __shared__ float part[8 * 16 * 16];            // per-wave partial tiles, 8 KB
    __shared__ float rowsum[16];

    const int tpb  = N_ / 16;
    const int b    = blockIdx.x / (H_ * tpb);
    const int rem  = blockIdx.x % (H_ * tpb);
    const int head = rem / tpb;
    const int i0   = (rem % tpb) * 16;

    const int t  = threadIdx.x;

    // kick off the async DMA of hw[b, :, head, :] -> hw_s before phase 1
    const float* hwb = hw + (size_t)b * N_ * D_ + head * DK_;
    for (int idx = t; idx < (N_ * 16) / 4; idx += 256) {
        int j = idx >> 2, q = idx & 3;              // row j, 16B chunk q
        copy16_g2lds(hwb + (size_t)j * D_ + q * 4, hw_s + idx * 4);
    }

    const int i  = t >> 4;                // row in tile, 0..15
    const int jg = t & 15;                // 16-wide j group

    const float* srcp = src + (size_t)(b * H_ + head) * N_;
    const float* dstp = dst + (size_t)(b * H_ + head) * N_;
    const float* arow = Aadj + (size_t)(b * N_ + i0 + i) * N_;

    const float sv = srcp[i0 + i];
    float m = -3.4e38f;
    float ev[16];
    #pragma unroll
    for (int q = 0; q < 16; ++q) {
        int   j = jg * 16 + q;
        float e = sv + dstp[j];
        e = (e >= 0.f) ? e : 0.2f * e;            // leaky_relu(., 0.2)
        e = (arow[j] > 0.f) ? e : NEG_;           // adjacency mask
        ev[q] = e;
        m = fmaxf(m, e);
    }
    // row max / sum across the 16 lanes owning this row (lanes 0-15 or 16-31)
    for (int msk = 1; msk < 16; msk <<= 1) m = fmaxf(m, __shfl_xor(m, msk, 16));
    float s = 0.f;
    #pragma unroll
    for (int q = 0; q < 16; ++q) {
        int   j = jg * 16 + q;
        float p = __expf(ev[q] - m);              // masked -> exp(NEG) == 0
        p_s[i * N_ + j] = p;
        s += p;
    }
    for (int msk = 1; msk < 16; msk <<= 1) s += __shfl_xor(s, msk, 16);
    if (jg == 0) rowsum[i] = s;

    async_wait_all();                     // hw_s DMA complete (this wave)
    __syncthreads();                      // all waves: p_s + hw_s visible

    // phase 2: each wave covers 32 j's (8 WMMA steps of K=4)
    const int lane = t & 31;
    const int wave = t >> 5;
    const int half = lane >> 4;
    const int l16  = lane & 15;

    v8f acc = {};
    #pragma unroll
    for (int kk = 0; kk < 8; ++kk) {
        int k0 = wave * 32 + kk * 4;
        v2f a, bf;
        a[0]  = p_s[l16 * N_ + k0 + 2 * half];
        a[1]  = p_s[l16 * N_ + k0 + 2 * half + 1];
        bf[0] = hw_s[(k0 + 2 * half)     * 16 + l16];
        bf[1] = hw_s[(k0 + 2 * half + 1) * 16 + l16];
        acc = __builtin_amdgcn_wmma_f32_16x16x4_f32(false, a, false, bf,
                                                    (short)0, acc, false, false);
    }
    #pragma unroll
    for (int r = 0; r < 8; ++r)
        part[(wave * 16 + r + 8 * half) * 16 + l16] = acc[r];
    __syncthreads();

    // reduce the 8 wave-partials, normalize, residual + ReLU, write h in place
    const int oi = t >> 4, od = t & 15;
    float v = 0.f;
    #pragma unroll
    for (int w = 0; w < 8; ++w) v += part[(w * 16 + oi) * 16 + od];
    v /= rowsum[oi];
    size_t gi = (size_t)(b * N_ + i0 + oi) * D_ + head * DK_ + od;
    v += h[gi];
    h[gi] = fmaxf(v, 0.f);
}

// ---------------------------------------------------------------------------
__global__ void pool_mean(const float* __restrict__ h, float* __restrict__ g)
{
    int tid = blockIdx.x * blockDim.x + threadIdx.x;    // B*D
    int b = tid / D_, d = tid % D_;
    float s = 0.f;
    for (int n = 0; n < N_; ++n) s += h[(size_t)(b * N_ + n) * D_ + d];
    g[tid] = s * (1.0f / N_);
}

__global__ void pred_head(const float* __restrict__ f1, const float* __restrict__ W,
                          const float* __restrict__ bias, float* __restrict__ out)
{
    int b = blockIdx.x * blockDim.x + threadIdx.x;
    if (b >= B_) return;
    float s = bias[0];
    for (int k = 0; k < FC_; ++k) s += f1[b * FC_ + k] * W[k];
    out[b] = s;
}

// ---------------------------------------------------------------------------
extern "C" void kernel_launch(void* const* d_in, const int* in_sizes, int n_in,
                              void* d_out, int out_size, void* d_ws, size_t ws_size,
                              hipStream_t stream)
{
    (void)in_sizes; (void)n_in; (void)out_size; (void)ws_size;
    const float* x      = (const float*)d_in[0];
    const float* Aadj   = (const float*)d_in[1];
    const float* emb_W  = (const float*)d_in[2];
    const float* emb_b  = (const float*)d_in[3];
    const float* gat_W  = (const float*)d_in[4];
    const float* gat_b  = (const float*)d_in[5];
    const float* gat_as = (const float*)d_in[6];
    const float* gat_ad = (const float*)d_in[7];
    const float* fc0_W  = (const float*)d_in[8];
    const float* fc0_b  = (const float*)d_in[9];
    const float* fc1_W  = (const float*)d_in[10];
    const float* fc1_b  = (const float*)d_in[11];
    const float* pred_W = (const float*)d_in[12];
    const float* pred_b = (const float*)d_in[13];
    float* out = (float*)d_out;

    float* ws  = (float*)d_ws;
    float* h   = ws;                           // B*N*D
    float* hw  = h   + (size_t)B_ * N_ * D_;   // B*N*D
    float* src = hw  + (size_t)B_ * N_ * D_;   // B*H*N
    float* dst = src + (size_t)B_ * H_ * N_;   // B*H*N
    float* g   = dst + (size_t)B_ * H_ * N_;   // B*D
    float* f0  = g   + (size_t)B_ * D_;        // B*FC
    float* f1  = f0  + (size_t)B_ * FC_;       // B*FC

    const int MR = B_ * N_;                    // 32768 node rows

    // h = x @ emb_W + emb_b
    gemm_f32_wmma<<<(MR / 16) * (D_ / 16) / 8, 256, 0, stream>>>(
        x, emb_W, emb_b, h, MR, F_, D_, 0);

    for (int l = 0; l < L_; ++l) {
        gemm_f32_wmma<<<(MR / 16) * (D_ / 16) / 8, 256, 0, stream>>>(
            h, gat_W + (size_t)l * D_ * D_, gat_b + l * D_, hw, MR, D_, D_, 0);
        gat_srcdst<<<(B_ * H_ * N_) / 256, 256, 0, stream>>>(
            hw, gat_as + l * H_ * DK_, gat_ad + l * H_ * DK_, src, dst);
        gat_attn<<<B_ * H_ * (N_ / 16), 256, 0, stream>>>(Aadj, hw, src, dst, h);
    }

    pool_mean<<<(B_ * D_) / 256, 256, 0, stream>>>(h, g);
    gemm_f32_wmma<<<(B_ / 16) * (FC_ / 16) / 8, 256, 0, stream>>>(
        g, fc0_W, fc0_b, f0, B_, D_, FC_, 1);
    gemm_f32_wmma<<<(B_ / 16) * (FC_ / 16) / 8, 256, 0, stream>>>(
        f0, fc1_W, fc1_b, f1, B_, FC_, FC_, 1);
    pred_head<<<1, 128, 0, stream>>>(f1, pred_W, pred_b, out);
}